// S6Layer_37374805409920
// MI455X (gfx1250) — compile-verified
//
#include <hip/hip_runtime.h>

typedef __attribute__((ext_vector_type(2))) float v2f;
typedef __attribute__((ext_vector_type(8))) float v8f;

#define NSTATE 16
#define CHUNK 16
#define NWAVE 16
#define THREADS (NWAVE * 32)
#define CPW 16                       // chunks per wave
#define STEPS_PER_WAVE (CHUNK * CPW) // 256
#define SEQ (NWAVE * STEPS_PER_WAVE) // 4096

// One workgroup per (batch, channel). Chunked-scan (SSD) formulation:
//   s_c      = G  x_c                (chunk state contribution, WMMA)
//   h_c      = weighted prefix-scan of s_c with decay Abar^16 (shuffles + LDS)
//   y_chunk  = M  x_c + H  h_c + D x (WMMA + epilogue)
__global__ __launch_bounds__(THREADS) void s6_wmma_kernel(
    const float* __restrict__ x, const float* __restrict__ A_log,
    const float* __restrict__ Bp, const float* __restrict__ Cp,
    const float* __restrict__ Dp, const float* __restrict__ dl,
    float* __restrict__ y, int d_model) {
  __shared__ float xs[SEQ];                 // 16 KB: whole sequence, this channel
  __shared__ float Gs[NSTATE][CHUNK];       // G[n][j] = Abar^(15-j) * Bbar
  __shared__ float Mm[CHUNK][CHUNK];        // M[j][s] = K[j-s] (lower-tri Toeplitz)
  __shared__ float Hm[CHUNK][NSTATE];       // H[j][n] = C_n * Abar^(j+1)
  __shared__ float pw[CHUNK + 1][NSTATE];   // Abar^k, k=0..16
  __shared__ float pc[NSTATE][CPW];         // (Abar^16)^c, c=0..15
  __shared__ float Bb[NSTATE], Cc[NSTATE];
  __shared__ float Ts[NWAVE][NSTATE];       // per-wave state totals
  __shared__ float Cr[NWAVE][NSTATE];       // cross-wave carries
  __shared__ float Hs[NWAVE][NSTATE][CPW];  // chunk-start states, B-operand staging

  const int tid = threadIdx.x;
  const int bd  = blockIdx.x;
  const int b   = bd / d_model, d = bd % d_model;
  const long xbase = (long)b * SEQ * d_model + d;

  // ---- stage x (t-strided; L2 merges the d-dimension across neighbor blocks)
  for (int i = tid; i < SEQ; i += THREADS)
    xs[i] = x[xbase + (long)i * d_model];

  // ---- per-state discretization + power tables
  if (tid < NSTATE) {
    const int n = tid;
    const float del  = dl[d];
    const float Abar = __expf(-del * __expf(A_log[d * NSTATE + n]));
    Bb[n] = del * Bp[d * NSTATE + n];
    Cc[n] = Cp[d * NSTATE + n];
    float p = 1.f;
    for (int k = 0; k <= CHUNK; ++k) { pw[k][n] = p; p *= Abar; }
    const float w16 = pw[CHUNK][n];
    p = 1.f;
    for (int c = 0; c < CPW; ++c) { pc[n][c] = p; p *= w16; }
  }
  __syncthreads();

  // ---- build G, H, M (one entry per thread)
  if (tid < 256) {
    const int r = tid >> 4, c2 = tid & 15;
    Gs[r][c2] = pw[15 - c2][r] * Bb[r];      // n=r, j=c2
    Hm[r][c2] = Cc[c2] * pw[r + 1][c2];      // j=r, n=c2
    float m = 0.f;
    if (c2 <= r) {
      const int tau = r - c2;
      for (int n = 0; n < NSTATE; ++n) m = fmaf(Cc[n] * pw[tau][n], Bb[n], m);
    }
    Mm[r][c2] = m;                           // j=r, s=c2
  }
  __syncthreads();

  const int wave = tid >> 5;
  const int lane = tid & 31;
  const int half = lane >> 4;   // which 16-lane half (ISA fragment layout)
  const int l16  = lane & 15;
  const int base = wave * STEPS_PER_WAVE;

  // ---- S = G * X  (rows = state n, cols = chunk c)
  v8f S = {};
#pragma unroll
  for (int k = 0; k < 4; ++k) {
    const int kk = 4 * k + 2 * half;
    v2f a, bf;
    a.x  = Gs[l16][kk];     a.y  = Gs[l16][kk + 1];              // A[M=l16][K]
    bf.x = xs[base + CHUNK * l16 + kk];                          // B[K][N=l16]
    bf.y = xs[base + CHUNK * l16 + kk + 1];
    S = __builtin_amdgcn_wmma_f32_16x16x4_f32(false, a, false, bf, (short)0, S,
                                              false, false);
  }

  // ---- weighted inclusive scan over the 16 chunk columns (per state row)
  float Ex[8];
#pragma unroll
  for (int r = 0; r < 8; ++r) {
    const int n = r + 8 * half;          // D layout: VGPR r holds M = r + 8*half
    float v  = S[r];
    float wd = pw[CHUNK][n];             // decay Abar^16
#pragma unroll
    for (int ds = 1; ds < 16; ds <<= 1) {
      const float up = __shfl_up(v, ds, 16);
      if (l16 >= ds) v = fmaf(wd, up, v);
      wd *= wd;
    }
    const float e = __shfl_up(v, 1, 16); // exclusive = chunk-start (local part)
    Ex[r] = (l16 == 0) ? 0.f : e;
    if (l16 == 15) Ts[wave][n] = v;      // wave total for cross-wave combine
  }
  __syncthreads();

  // ---- cross-wave carry: Horner scan with decay W = Abar^256
  if (tid < 256) {
    const int wv = tid >> 4, n = tid & 15;
    const float W = pc[n][15] * pw[CHUNK][n];
    float cr = 0.f;
    for (int wp = 0; wp < wv; ++wp) cr = fmaf(cr, W, Ts[wp][n]);
    Cr[wv][n] = cr;
  }
  __syncthreads();

  // ---- chunk-start states -> LDS in B-operand orientation
#pragma unroll
  for (int r = 0; r < 8; ++r) {
    const int n = r + 8 * half;
    Hs[wave][n][l16] = Ex[r] + pc[n][l16] * Cr[wave][n];
  }
  __syncthreads();

  // ---- Y = M*X + H*Hstart
  v8f Y = {};
#pragma unroll
  for (int k = 0; k < 4; ++k) {
    const int kk = 4 * k + 2 * half;
    v2f a, bf;
    a.x  = Mm[l16][kk];     a.y  = Mm[l16][kk + 1];
    bf.x = xs[base + CHUNK * l16 + kk];
    bf.y = xs[base + CHUNK * l16 + kk + 1];
    Y = __builtin_amdgcn_wmma_f32_16x16x4_f32(false, a, false, bf, (short)0, Y,
                                              false, false);
  }
#pragma unroll
  for (int k = 0; k < 4; ++k) {
    const int kk = 4 * k + 2 * half;
    v2f a, bf;
    a.x  = Hm[l16][kk];     a.y  = Hm[l16][kk + 1];
    bf.x = Hs[wave][kk][l16];
    bf.y = Hs[wave][kk + 1][l16];
    Y = __builtin_amdgcn_wmma_f32_16x16x4_f32(false, a, false, bf, (short)0, Y,
                                              false, false);
  }

  // ---- epilogue: + D*x, scatter store (L2 write-combines across blocks)
  const float Dd = Dp[d];
#pragma unroll
  for (int r = 0; r < 8; ++r) {
    const int j = r + 8 * half;                  // position in chunk
    const int t = base + CHUNK * l16 + j;        // absolute timestep
    y[xbase + (long)t * d_model] = Y[r] + Dd * xs[t];
  }
}

extern "C" void kernel_launch(void* const* d_in, const int* in_sizes, int n_in,
                              void* d_out, int out_size, void* d_ws, size_t ws_size,
                              hipStream_t stream) {
  const float* x     = (const float*)d_in[0];
  const float* A_log = (const float*)d_in[1];
  const float* B     = (const float*)d_in[2];
  const float* C     = (const float*)d_in[3];
  const float* D     = (const float*)d_in[4];
  const float* delta = (const float*)d_in[5];
  float* y = (float*)d_out;

  const int d_model = in_sizes[4];                      // |D| = d_model
  const long total  = (long)in_sizes[0];                // batch*seq*d_model
  const int batch   = (int)(total / ((long)d_model * SEQ));

  dim3 grid(batch * d_model), block(THREADS);
  hipLaunchKernelGGL(s6_wmma_kernel, grid, block, 0, stream,
                     x, A_log, B, C, D, delta, y, d_model);
}